// PaiNNTLScore_10187662426869
// MI455X (gfx1250) — compile-verified
//
#include <hip/hip_runtime.h>
#include <math.h>

typedef __attribute__((ext_vector_type(2))) float v2f;
typedef __attribute__((ext_vector_type(8))) float v8f;

#define PI_OVER_LEN 0.31415926535897932f /* pi / LEN, LEN=10 */
#define WPB 8                            /* waves per block (256 threads) */

__device__ __forceinline__ v2f mkv2(float a, float b) { v2f r; r.x = a; r.y = b; return r; }

__device__ __forceinline__ v8f wmma4(v2f a, v2f b, v8f c) {
  // D = A(16x4 f32) * B(4x16 f32) + C(16x16 f32)
  return __builtin_amdgcn_wmma_f32_16x16x4_f32(false, a, false, b, (short)0, c, false, false);
}

__device__ __forceinline__ float silu(float x) { return x / (1.0f + __expf(-x)); }

// B fragment: 4x16 tile of row-major weight w (K x Nout); chunk c = rows 4c..4c+3,
// tile t = cols 16t..16t+15. Lanes 0-15: K=4c+0/1, lanes 16-31: K=4c+2/3 (koff).
__device__ __forceinline__ v2f bfrag(const float* __restrict__ w, int Nout,
                                     int c, int t, int koff, int nn) {
  const float* p = w + (4 * c + koff) * Nout + (nn + 16 * t);
  return mkv2(p[0], p[Nout]);
}

__device__ __forceinline__ void atomic_addf(float* p, float v) {
  __hip_atomic_fetch_add(p, v, __ATOMIC_RELAXED, __HIP_MEMORY_SCOPE_AGENT);
}

// ---------------------------------------------------------------------------
// Edge message kernel: one wave per 16 edges. Fused: posenc -> two MLPs
// (WMMA f32) -> mix -> cross-product message -> atomic scatter into s1/v1.
// ---------------------------------------------------------------------------
__global__ void __launch_bounds__(256) painn_edge_kernel(
    const float* __restrict__ s0, const float* __restrict__ v0,
    float* __restrict__ s1, float* __restrict__ v1,
    const int* __restrict__ ei, const float* __restrict__ dist,
    const float* __restrict__ dirp,
    const float* __restrict__ mpw1, const float* __restrict__ mpb1,
    const float* __restrict__ mpw2, const float* __restrict__ mpb2,
    const float* __restrict__ mww1, const float* __restrict__ mwb1,
    const float* __restrict__ mww2, const float* __restrict__ mwb2,
    int Ee) {
  __shared__ float hS[WPB][16][32];
  __shared__ float hP[WPB][16][32];
  __shared__ int esrc[WPB][16];
  __shared__ int edst[WPB][16];

  const int lane = threadIdx.x & 31;
  const int wv = threadIdx.x >> 5;
  const int tile = blockIdx.x * WPB + wv;
  const int e0 = tile * 16;
  const bool active = (e0 < Ee);
  const int m = lane & 15;            // row (edge) for A, col (N) for B/D
  const int nn = m;
  const int koff = (lane >> 4) << 1;  // 0 for lanes 0-15, 2 for lanes 16-31
  const int mhi = (lane >> 4) << 3;   // D-layout row offset

  if (active && lane < 16) {
    esrc[wv][lane] = ei[e0 + lane];
    edst[wv][lane] = ei[Ee + e0 + lane];
  }
  __syncthreads();

  if (active) {
    const int rowS = esrc[wv][m];
    const float de = dist[e0 + m];
    v2f aS[8], aP[8];
#pragma unroll
    for (int c = 0; c < 8; ++c) {
      const float* p = s0 + rowS * 32 + 4 * c + koff;
      aS[c] = mkv2(p[0], p[1]);
#pragma unroll
      for (int q = 0; q < 2; ++q) {
        const int k = 4 * c + koff + q;
        const float ang = de * (float)((k & 15) + 1) * PI_OVER_LEN;
        aP[c][q] = (k < 16) ? __sinf(ang) : __cosf(ang);
      }
    }
    // first layers of both MLPs: 16x32 @ 32x32
    v8f s_h0 = {}, s_h1 = {}, p_h0 = {}, p_h1 = {};
#pragma unroll
    for (int c = 0; c < 8; ++c) {
      s_h0 = wmma4(aS[c], bfrag(mpw1, 32, c, 0, koff, nn), s_h0);
      s_h1 = wmma4(aS[c], bfrag(mpw1, 32, c, 1, koff, nn), s_h1);
      p_h0 = wmma4(aP[c], bfrag(mww1, 32, c, 0, koff, nn), p_h0);
      p_h1 = wmma4(aP[c], bfrag(mww1, 32, c, 1, koff, nn), p_h1);
    }
    const float sb0 = mpb1[nn], sb1 = mpb1[nn + 16];
    const float pb0 = mwb1[nn], pb1 = mwb1[nn + 16];
#pragma unroll
    for (int j = 0; j < 8; ++j) {  // D-layout -> row-major staging in LDS
      hS[wv][j + mhi][nn] = silu(s_h0[j] + sb0);
      hS[wv][j + mhi][nn + 16] = silu(s_h1[j] + sb1);
      hP[wv][j + mhi][nn] = silu(p_h0[j] + pb0);
      hP[wv][j + mhi][nn + 16] = silu(p_h1[j] + pb1);
    }
  }
  __syncthreads();

  if (active) {
    // second layers: 16x32 @ 32x128, then elementwise mix
    v8f mixT[8];
#pragma unroll
    for (int t = 0; t < 8; ++t) {
      v8f accS = {}, accP = {};
#pragma unroll
      for (int c = 0; c < 8; ++c) {
        const v2f ah = mkv2(hS[wv][m][4 * c + koff], hS[wv][m][4 * c + koff + 1]);
        const v2f ap = mkv2(hP[wv][m][4 * c + koff], hP[wv][m][4 * c + koff + 1]);
        accS = wmma4(ah, bfrag(mpw2, 128, c, t, koff, nn), accS);
        accP = wmma4(ap, bfrag(mww2, 128, c, t, koff, nn), accP);
      }
      const float b2s = mpb2[nn + 16 * t];
      const float b2p = mwb2[nn + 16 * t];
#pragma unroll
      for (int j = 0; j < 8; ++j) mixT[t][j] = (accS[j] + b2s) * (accP[j] + b2p);
    }
    // split: gates=t0/1, cpg=t2/3, sed=t4/5, sf=t6/7 ; scatter messages
#pragma unroll
    for (int j = 0; j < 8; ++j) {
      const int mr = j + mhi;
      const int sE = esrc[wv][mr];
      const int dE = edst[wv][mr];
      const int eg = e0 + mr;
      const float dx = dirp[3 * eg + 0];
      const float dy = dirp[3 * eg + 1];
      const float dz = dirp[3 * eg + 2];
#pragma unroll
      for (int tp = 0; tp < 2; ++tp) {
        const int f = nn + 16 * tp;
        const float gat = (tp == 0) ? mixT[0][j] : mixT[1][j];
        const float cpg = (tp == 0) ? mixT[2][j] : mixT[3][j];
        const float sed = (tp == 0) ? mixT[4][j] : mixT[5][j];
        const float sfv = (tp == 0) ? mixT[6][j] : mixT[7][j];
        const float vdx = v0[dE * 96 + f * 3 + 0];
        const float vdy = v0[dE * 96 + f * 3 + 1];
        const float vdz = v0[dE * 96 + f * 3 + 2];
        const float vsx = v0[sE * 96 + f * 3 + 0];
        const float vsy = v0[sE * 96 + f * 3 + 1];
        const float vsz = v0[sE * 96 + f * 3 + 2];
        const float cx = dy * vdz - dz * vdy;  // dir x v[dst]
        const float cy = dz * vdx - dx * vdz;
        const float cz = dx * vdy - dy * vdx;
        atomic_addf(&v1[dE * 96 + f * 3 + 0], sed * dx + gat * vsx + cpg * cx);
        atomic_addf(&v1[dE * 96 + f * 3 + 1], sed * dy + gat * vsy + cpg * cy);
        atomic_addf(&v1[dE * 96 + f * 3 + 2], sed * dz + gat * vsz + cpg * cz);
        atomic_addf(&s1[dE * 32 + f], sfv * s0[sE * 32 + f]);
      }
    }
  }
}

// ---------------------------------------------------------------------------
// Node update kernel: one wave per 16 nodes. u_V/u_U channel matmuls, norm,
// 64->32->96 MLP, elementwise update. Reads s1/v1 (accumulated), writes s0/v0.
// ---------------------------------------------------------------------------
__global__ void __launch_bounds__(256) painn_node_kernel(
    const float* __restrict__ s1, const float* __restrict__ v1,
    float* __restrict__ s0, float* __restrict__ v0,
    const float* __restrict__ uU, const float* __restrict__ uV,
    const float* __restrict__ uw1, const float* __restrict__ ub1,
    const float* __restrict__ uw2, const float* __restrict__ ub2,
    int Nn) {
  __shared__ float vnsh[WPB][16][32];
  __shared__ float hsh[WPB][16][32];

  const int lane = threadIdx.x & 31;
  const int wv = threadIdx.x >> 5;
  const int tile = blockIdx.x * WPB + wv;
  const int n0 = tile * 16;
  const bool active = (n0 < Nn);
  const int m = lane & 15;
  const int nn = m;
  const int koff = (lane >> 4) << 1;
  const int mhi = (lane >> 4) << 3;

  v8f Uv[2][3] = {};
  v8f Vn2[2] = {};
  v8f go[2] = {}, ssn[2] = {}, addv[2] = {};

  if (active) {
    const int row = n0 + m;
    v8f Vv[2][3] = {};
#pragma unroll
    for (int cc = 0; cc < 3; ++cc) {
#pragma unroll
      for (int c = 0; c < 8; ++c) {
        const v2f a = mkv2(v1[row * 96 + (4 * c + koff) * 3 + cc],
                           v1[row * 96 + (4 * c + koff + 1) * 3 + cc]);
#pragma unroll
        for (int t = 0; t < 2; ++t) {
          Vv[t][cc] = wmma4(a, bfrag(uV, 32, c, t, koff, nn), Vv[t][cc]);
          Uv[t][cc] = wmma4(a, bfrag(uU, 32, c, t, koff, nn), Uv[t][cc]);
        }
      }
    }
#pragma unroll
    for (int t = 0; t < 2; ++t) {
#pragma unroll
      for (int j = 0; j < 8; ++j) {
        const float x = Vv[t][0][j], y = Vv[t][1][j], z = Vv[t][2][j];
        const float d2 = x * x + y * y + z * z;
        Vn2[t][j] = d2;
        vnsh[wv][j + mhi][nn + 16 * t] = sqrtf(d2);
      }
    }
  }
  __syncthreads();

  if (active) {
    const int row = n0 + m;
    v8f h0 = {}, h1 = {};
#pragma unroll
    for (int c = 0; c < 16; ++c) {  // K=64: [Vn | s]
      v2f a;
      if (c < 8) {
        a = mkv2(vnsh[wv][m][4 * c + koff], vnsh[wv][m][4 * c + koff + 1]);
      } else {
        const float* p = s1 + row * 32 + 4 * (c - 8) + koff;
        a = mkv2(p[0], p[1]);
      }
      h0 = wmma4(a, bfrag(uw1, 32, c, 0, koff, nn), h0);
      h1 = wmma4(a, bfrag(uw1, 32, c, 1, koff, nn), h1);
    }
    const float b0 = ub1[nn], b1 = ub1[nn + 16];
#pragma unroll
    for (int j = 0; j < 8; ++j) {
      hsh[wv][j + mhi][nn] = silu(h0[j] + b0);
      hsh[wv][j + mhi][nn + 16] = silu(h1[j] + b1);
    }
  }
  __syncthreads();

  if (active) {
#pragma unroll
    for (int t = 0; t < 6; ++t) {  // 16x32 @ 32x96 -> g|ssn|add
      v8f acc = {};
#pragma unroll
      for (int c = 0; c < 8; ++c) {
        const v2f a = mkv2(hsh[wv][m][4 * c + koff], hsh[wv][m][4 * c + koff + 1]);
        acc = wmma4(a, bfrag(uw2, 96, c, t, koff, nn), acc);
      }
      const float bb = ub2[nn + 16 * t];
#pragma unroll
      for (int j = 0; j < 8; ++j) acc[j] += bb;
      if (t < 2) go[t] = acc;
      else if (t < 4) ssn[t - 2] = acc;
      else addv[t - 4] = acc;
    }
#pragma unroll
    for (int j = 0; j < 8; ++j) {
      const int nd = n0 + j + mhi;
#pragma unroll
      for (int tp = 0; tp < 2; ++tp) {
        const int f = nn + 16 * tp;
        s0[nd * 32 + f] = s1[nd * 32 + f] + Vn2[tp][j] * ssn[tp][j] + addv[tp][j];
        const float g = go[tp][j];
#pragma unroll
        for (int cc = 0; cc < 3; ++cc)
          v0[nd * 96 + f * 3 + cc] = v1[nd * 96 + f * 3 + cc] + g * Uv[tp][cc][j];
      }
    }
  }
}

// ---------------------------------------------------------------------------
// Readout: MLP(s) -> [inv_out | gates]; eqv_out = gates * (v @ ro_V).
// d_out = [inv_out (N*32) | eqv_out (N*96)] fp32.
// ---------------------------------------------------------------------------
__global__ void __launch_bounds__(256) painn_readout_kernel(
    const float* __restrict__ s0, const float* __restrict__ v0,
    const float* __restrict__ rw1, const float* __restrict__ rb1,
    const float* __restrict__ rw2, const float* __restrict__ rb2,
    const float* __restrict__ rV, float* __restrict__ out, int Nn) {
  __shared__ float hsh[WPB][16][32];

  const int lane = threadIdx.x & 31;
  const int wv = threadIdx.x >> 5;
  const int tile = blockIdx.x * WPB + wv;
  const int n0 = tile * 16;
  const bool active = (n0 < Nn);
  const int m = lane & 15;
  const int nn = m;
  const int koff = (lane >> 4) << 1;
  const int mhi = (lane >> 4) << 3;

  if (active) {
    const int row = n0 + m;
    v8f h0 = {}, h1 = {};
#pragma unroll
    for (int c = 0; c < 8; ++c) {
      const float* p = s0 + row * 32 + 4 * c + koff;
      const v2f a = mkv2(p[0], p[1]);
      h0 = wmma4(a, bfrag(rw1, 32, c, 0, koff, nn), h0);
      h1 = wmma4(a, bfrag(rw1, 32, c, 1, koff, nn), h1);
    }
    const float b0 = rb1[nn], b1 = rb1[nn + 16];
#pragma unroll
    for (int j = 0; j < 8; ++j) {
      hsh[wv][j + mhi][nn] = silu(h0[j] + b0);
      hsh[wv][j + mhi][nn + 16] = silu(h1[j] + b1);
    }
  }
  __syncthreads();

  if (active) {
    const int row = n0 + m;
    v8f outT[4];
#pragma unroll
    for (int t = 0; t < 4; ++t) {  // 16x32 @ 32x64 -> inv|gates
      v8f acc = {};
#pragma unroll
      for (int c = 0; c < 8; ++c) {
        const v2f a = mkv2(hsh[wv][m][4 * c + koff], hsh[wv][m][4 * c + koff + 1]);
        acc = wmma4(a, bfrag(rw2, 64, c, t, koff, nn), acc);
      }
      const float bb = rb2[nn + 16 * t];
#pragma unroll
      for (int j = 0; j < 8; ++j) acc[j] += bb;
      outT[t] = acc;
    }
    v8f Vv[2][3] = {};
#pragma unroll
    for (int cc = 0; cc < 3; ++cc) {
#pragma unroll
      for (int c = 0; c < 8; ++c) {
        const v2f a = mkv2(v0[row * 96 + (4 * c + koff) * 3 + cc],
                           v0[row * 96 + (4 * c + koff + 1) * 3 + cc]);
#pragma unroll
        for (int t = 0; t < 2; ++t)
          Vv[t][cc] = wmma4(a, bfrag(rV, 32, c, t, koff, nn), Vv[t][cc]);
      }
    }
#pragma unroll
    for (int j = 0; j < 8; ++j) {
      const int nd = n0 + j + mhi;
#pragma unroll
      for (int tp = 0; tp < 2; ++tp) {
        const int f = nn + 16 * tp;
        out[nd * 32 + f] = outT[tp][j];
        const float g = outT[2 + tp][j];
#pragma unroll
        for (int cc = 0; cc < 3; ++cc)
          out[Nn * 32 + nd * 96 + f * 3 + cc] = g * Vv[tp][cc][j];
      }
    }
  }
}

// ---------------------------------------------------------------------------
extern "C" void kernel_launch(void* const* d_in, const int* in_sizes, int n_in,
                              void* d_out, int out_size, void* d_ws, size_t ws_size,
                              hipStream_t stream) {
  const float* s_in = (const float*)d_in[0];
  const float* v_in = (const float*)d_in[1];
  const int* ei = (const int*)d_in[2];
  const float* dist = (const float*)d_in[3];
  const float* dirp = (const float*)d_in[4];
  const float* mp_w1 = (const float*)d_in[5];
  const float* mp_b1 = (const float*)d_in[6];
  const float* mp_w2 = (const float*)d_in[7];
  const float* mp_b2 = (const float*)d_in[8];
  const float* mw_w1 = (const float*)d_in[9];
  const float* mw_b1 = (const float*)d_in[10];
  const float* mw_w2 = (const float*)d_in[11];
  const float* mw_b2 = (const float*)d_in[12];
  const float* u_U = (const float*)d_in[13];
  const float* u_V = (const float*)d_in[14];
  const float* u_w1 = (const float*)d_in[15];
  const float* u_b1 = (const float*)d_in[16];
  const float* u_w2 = (const float*)d_in[17];
  const float* u_b2 = (const float*)d_in[18];
  const float* ro_w1 = (const float*)d_in[19];
  const float* ro_b1 = (const float*)d_in[20];
  const float* ro_w2 = (const float*)d_in[21];
  const float* ro_b2 = (const float*)d_in[22];
  const float* ro_V = (const float*)d_in[23];

  const int Nn = in_sizes[0] / 32;  // 50000
  const int Ee = in_sizes[3];       // 800000

  float* ws = (float*)d_ws;
  float* s0 = ws;                          // N*32
  float* v0 = ws + (size_t)Nn * 32;        // N*96
  float* s1 = ws + (size_t)Nn * 128;       // N*32
  float* v1 = s1 + (size_t)Nn * 32;        // N*96

  hipMemcpyAsync(s0, s_in, (size_t)Nn * 32 * sizeof(float), hipMemcpyDeviceToDevice, stream);
  hipMemcpyAsync(v0, v_in, (size_t)Nn * 96 * sizeof(float), hipMemcpyDeviceToDevice, stream);

  const int tilesE = (Ee + 15) / 16;
  const int tilesN = (Nn + 15) / 16;
  const dim3 blk(256);
  const dim3 gridE((tilesE + WPB - 1) / WPB);
  const dim3 gridN((tilesN + WPB - 1) / WPB);

  for (int l = 0; l < 2; ++l) {
    // seed accumulators with current state: s1|v1 <- s0|v0 (contiguous N*128)
    hipMemcpyAsync(s1, s0, (size_t)Nn * 128 * sizeof(float), hipMemcpyDeviceToDevice, stream);
    painn_edge_kernel<<<gridE, blk, 0, stream>>>(
        s0, v0, s1, v1, ei, dist, dirp,
        mp_w1 + (size_t)l * 32 * 32, mp_b1 + (size_t)l * 32,
        mp_w2 + (size_t)l * 32 * 128, mp_b2 + (size_t)l * 128,
        mw_w1 + (size_t)l * 32 * 32, mw_b1 + (size_t)l * 32,
        mw_w2 + (size_t)l * 32 * 128, mw_b2 + (size_t)l * 128, Ee);
    painn_node_kernel<<<gridN, blk, 0, stream>>>(
        s1, v1, s0, v0,
        u_U + (size_t)l * 32 * 32, u_V + (size_t)l * 32 * 32,
        u_w1 + (size_t)l * 64 * 32, u_b1 + (size_t)l * 32,
        u_w2 + (size_t)l * 32 * 96, u_b2 + (size_t)l * 96, Nn);
  }
  painn_readout_kernel<<<gridN, blk, 0, stream>>>(
      s0, v0, ro_w1, ro_b1, ro_w2, ro_b2, ro_V, (float*)d_out, Nn);
}